// EAdjective_28295244546074
// MI455X (gfx1250) — compile-verified
//
#include <hip/hip_runtime.h>
#include <hip/hip_bf16.h>

typedef __attribute__((ext_vector_type(2))) float v2f;
typedef __attribute__((ext_vector_type(8))) float v8f;

#define NROWS 2048   // N == M
#define DIM   64     // in/out feature dim
#define WIDTH 5      // hidden width

// ---------------------------------------------------------------------------
// Kernel 1: shared-weight MLP for all 4096 rows (2048 of x, 2048 of y).
// Layer 0 ([rows,64] @ [64,5]^T, padded to N=16) runs on V_WMMA_F32_16X16X4_F32;
// the 5x5 hidden layers and 5->64 output layer are trivial VALU per row.
// Block = 128 threads (4 waves); each wave owns a 16-row tile for WMMA.
// ---------------------------------------------------------------------------
__global__ void __launch_bounds__(128)
mlp_wmma_kernel(const float* __restrict__ x, const float* __restrict__ y,
                const float* __restrict__ W0, const float* __restrict__ b0,
                const float* __restrict__ Wh, const float* __restrict__ bh,
                const float* __restrict__ Wout, const float* __restrict__ bout,
                float* __restrict__ t /* [4096][64] in d_ws */)
{
    __shared__ float hbuf[64 * WIDTH];   // layer-0 activations for this block's 64 rows

    const int tid  = threadIdx.x;
    const int lane = tid & 31;
    const int wave = tid >> 5;

    // Global row tile for this wave (rows 0..4095; >=2048 means source y)
    const int rowTile = blockIdx.x * 64 + wave * 16;
    const bool isY    = rowTile >= NROWS;
    const float* src  = isY ? y : x;
    const int srcRow  = rowTile - (isY ? NROWS : 0);

    // WMMA fragment coordinates (wave32, 16x16x4 f32):
    //   A: lanes 0-15 hold M=lane, VGPR0/1 = K={0,1}; lanes 16-31 K={2,3}
    //   B: lanes hold N=lane&15 with the same K split
    const int m  = lane & 15;        // M (A) / N (B)
    const int hi = lane >> 4;        // 0 -> K{0,1}, 1 -> K{2,3}

    const float* arow = src + (size_t)(srcRow + m) * DIM + hi * 2;
    // Clamp padded lanes (n >= WIDTH) to row 0 so the load is always valid;
    // zero them with a multiply (no exec-mask predication, no branches).
    const float* brow = W0 + (m < WIDTH ? m : 0) * DIM + hi * 2;
    const float  bmask = (m < WIDTH) ? 1.0f : 0.0f;

    // Hoist all 16 B fragments (W0 is tiny and reused by every tile).
    v2f bfrag[16];
    #pragma unroll
    for (int s = 0; s < 16; ++s) {
        v2f bv = *(const v2f*)(brow + s * 4);   // 8B-aligned -> global_load_b64
        bfrag[s].x = bv.x * bmask;
        bfrag[s].y = bv.y * bmask;
    }

    v8f acc = {};
    #pragma unroll
    for (int s = 0; s < 16; ++s) {
        v2f a = *(const v2f*)(arow + s * 4);    // 8B-aligned -> global_load_b64
        acc = __builtin_amdgcn_wmma_f32_16x16x4_f32(
                  false, a, false, bfrag[s], (short)0, acc, false, false);
    }

    // D layout: acc[v] = h[rowTile + hi*8 + v][n], n = m.  Bias + ReLU -> LDS.
    if (m < WIDTH) {
        const float bias = b0[m];
        #pragma unroll
        for (int v = 0; v < 8; ++v) {
            float hv = acc[v] + bias;
            hv = fmaxf(hv, 0.0f);
            hbuf[(wave * 16 + hi * 8 + v) * WIDTH + m] = hv;
        }
    }
    __syncthreads();

    // Tail: one thread per row finishes the 3 shared hidden layers + output layer.
    if (tid < 64) {
        float h[WIDTH], hn[WIDTH];
        #pragma unroll
        for (int q = 0; q < WIDTH; ++q) h[q] = hbuf[tid * WIDTH + q];

        for (int l = 0; l < 3; ++l) {               // DEPTH-2 shared hidden layers
            #pragma unroll
            for (int n = 0; n < WIDTH; ++n) {
                float s = bh[n];
                #pragma unroll
                for (int q = 0; q < WIDTH; ++q) s += h[q] * Wh[n * WIDTH + q];
                hn[n] = fmaxf(s, 0.0f);
            }
            #pragma unroll
            for (int n = 0; n < WIDTH; ++n) h[n] = hn[n];
        }

        const int grow = blockIdx.x * 64 + tid;     // 0..4095
        float* trow = t + (size_t)grow * DIM;
        #pragma unroll 8
        for (int c = 0; c < DIM; ++c) {
            float s = bout[c];
            #pragma unroll
            for (int q = 0; q < WIDTH; ++q) s += h[q] * Wout[c * WIDTH + q];
            trow[c] = fmaxf(s, 0.0f);
        }
    }
}

// ---------------------------------------------------------------------------
// Kernel 2: out[i][j] = sum_k relu(ty[j][k] - tx[i][k])^2.
// ReLU inside the reduction => not expressible as GEMM; register-blocked VALU.
// 64x64 tile per 256-thread block; 4x4 micro-tile per thread; LDS stride 65
// keeps the k-indexed b32 reads bank-conflict free.
// ---------------------------------------------------------------------------
#define LSTRIDE 65

__global__ void __launch_bounds__(256)
pairwise_kernel(const float* __restrict__ t, float* __restrict__ out)
{
    __shared__ float txs[64 * LSTRIDE];
    __shared__ float tys[64 * LSTRIDE];

    const float* tx = t;                       // rows 0..2047  (from x)
    const float* ty = t + (size_t)NROWS * DIM; // rows 2048..4095 (from y)

    const int i0  = blockIdx.y * 64;
    const int j0  = blockIdx.x * 64;
    const int tid = threadIdx.x;

    // Stage tiles: 4096 floats each, coalesced b32 loads (consecutive tid -> k).
    for (int e = tid; e < 64 * DIM; e += 256) {
        const int r = e >> 6;
        const int c = e & 63;
        txs[r * LSTRIDE + c] = tx[(size_t)(i0 + r) * DIM + c];
        tys[r * LSTRIDE + c] = ty[(size_t)(j0 + r) * DIM + c];
    }
    __syncthreads();

    const int ti = tid >> 4;   // 0..15 -> 4 rows of i
    const int tj = tid & 15;   // 0..15 -> 4 cols of j
    const float* txr = &txs[(ti * 4) * LSTRIDE];
    const float* tyr = &tys[(tj * 4) * LSTRIDE];

    float acc[4][4];
    #pragma unroll
    for (int r = 0; r < 4; ++r)
        #pragma unroll
        for (int c = 0; c < 4; ++c) acc[r][c] = 0.0f;

    #pragma unroll 4
    for (int k = 0; k < DIM; ++k) {
        float av[4], bv[4];
        #pragma unroll
        for (int r = 0; r < 4; ++r) av[r] = txr[r * LSTRIDE + k];
        #pragma unroll
        for (int c = 0; c < 4; ++c) bv[c] = tyr[c * LSTRIDE + k];
        #pragma unroll
        for (int r = 0; r < 4; ++r) {
            #pragma unroll
            for (int c = 0; c < 4; ++c) {
                float d = fmaxf(bv[c] - av[r], 0.0f);
                acc[r][c] = fmaf(d, d, acc[r][c]);
            }
        }
    }

    #pragma unroll
    for (int r = 0; r < 4; ++r) {
        float* orow = out + (size_t)(i0 + ti * 4 + r) * NROWS + j0 + tj * 4;
        #pragma unroll
        for (int c = 0; c < 4; ++c) orow[c] = acc[r][c];
    }
}

extern "C" void kernel_launch(void* const* d_in, const int* in_sizes, int n_in,
                              void* d_out, int out_size, void* d_ws, size_t ws_size,
                              hipStream_t stream) {
    const float* x    = (const float*)d_in[0];
    const float* y    = (const float*)d_in[1];
    const float* W0   = (const float*)d_in[2];
    const float* b0   = (const float*)d_in[3];
    const float* Wh   = (const float*)d_in[4];
    const float* bh   = (const float*)d_in[5];
    const float* Wout = (const float*)d_in[6];
    const float* bout = (const float*)d_in[7];
    float* out = (float*)d_out;
    float* t   = (float*)d_ws;   // [4096][64] = 1 MB scratch

    // Phase 1: MLP (WMMA layer-0), 4096 rows, 64 rows per block.
    mlp_wmma_kernel<<<64, 128, 0, stream>>>(x, y, W0, b0, Wh, bh, Wout, bout, t);

    // Phase 2: pairwise masked squared distance, 64x64 tile per block.
    pairwise_kernel<<<dim3(32, 32), 256, 0, stream>>>(t, out);
}